// CapsuleLayer_8375186227255
// MI455X (gfx1250) — compile-verified
//
#include <hip/hip_runtime.h>
#include <math.h>

// ---------------------------------------------------------------------------
// CapsNet dynamic routing for MI455X (gfx1250).
// One workgroup = (capsule c, tile of 16 batches). Priors are recomputed from
// L2 via V_WMMA_F32_16X16X4_F32 each pass (5 passes total); routing logits
// live in LDS. Nothing large ever touches HBM.
// ---------------------------------------------------------------------------

typedef float v2f __attribute__((ext_vector_type(2)));
typedef float v8f __attribute__((ext_vector_type(8)));

// Probe the builtin ONLY in the device pass; host clang never has it.
#if defined(__HIP_DEVICE_COMPILE__)
#if !__has_builtin(__builtin_amdgcn_wmma_f32_16x16x4_f32)
#error "device: missing __builtin_amdgcn_wmma_f32_16x16x4_f32"
#endif
#endif

#define R_     1152
#define CIN_   8
#define COUT_  16
#define B_     256
#define NCAPS_ 10
#define BT_    16          // batch tile = WMMA M
#define WAVES_ 8           // 256 threads
#define RPW_   (R_ / WAVES_)   // 144 route nodes per wave per pass

__device__ __forceinline__ v8f zero8() {
    v8f z;
#pragma unroll
    for (int j = 0; j < 8; ++j) z[j] = 0.0f;
    return z;
}

// One K=8 prior-tile matmul for route node r:
//   D[b,o] (+)= sum_i (pscale * x[b,r,i]) * W[c,r,i,o]
// A layout (32-bit 16x4): lanes 0-15 hold M=0..15 / K={k0,k0+1}, lanes 16-31
// the upper K pair. B layout mirrors it with N across lanes.
__device__ __forceinline__ v8f wmma_k8(const float* __restrict__ xp,
                                       const float* __restrict__ wp,
                                       int half, int bn, float pscale, v8f acc)
{
#if defined(__HIP_DEVICE_COMPILE__)
#pragma unroll
    for (int kk = 0; kk < 2; ++kk) {
        const int k0 = kk * 4 + half * 2;
        v2f a;
        a.x = xp[k0]     * pscale;
        a.y = xp[k0 + 1] * pscale;
        v2f b;
        b.x = wp[k0 * COUT_ + bn];
        b.y = wp[k0 * COUT_ + COUT_ + bn];
        acc = __builtin_amdgcn_wmma_f32_16x16x4_f32(
            /*neg_a=*/false, a, /*neg_b=*/false, b,
            /*c_mod=*/(short)0, acc, /*reuse_a=*/false, /*reuse_b=*/false);
    }
#endif
    return acc;
}

// Reduce per-wave WMMA accumulators across the 8 waves, apply squash, and
// publish the 16x16 output tile to lds_out[m*16+n].
__device__ __forceinline__ void reduce_squash(v8f acc,
                                              float* __restrict__ lds_red,
                                              float* __restrict__ lds_out,
                                              int tid, int wv, int half, int bn)
{
    __syncthreads();
#pragma unroll
    for (int j = 0; j < 8; ++j)
        lds_red[wv * 256 + (j + 8 * half) * 16 + bn] = acc[j];   // D layout -> [m][n]
    __syncthreads();
    // thread tid owns (m = tid>>4, n = tid&15)
    float s = 0.0f;
#pragma unroll
    for (int w = 0; w < WAVES_; ++w) s += lds_red[w * 256 + tid];
    // squash: out = s * sqrt(q)/(1+q), q = sum_n s^2 (16 consecutive lanes)
    float q = s * s;
    q += __shfl_xor(q, 1, 32);
    q += __shfl_xor(q, 2, 32);
    q += __shfl_xor(q, 4, 32);
    q += __shfl_xor(q, 8, 32);
    const float sc = __fsqrt_rn(q) / (1.0f + q);
    lds_out[tid] = s * sc;
    __syncthreads();
}

__global__ __launch_bounds__(256)
void caps_route_wmma(const float* __restrict__ x,     // [B][R][CIN]
                     const float* __restrict__ rw,    // [NCAPS][R][CIN][COUT]
                     float* __restrict__ out)         // [NCAPS][B][COUT]
{
    __shared__ float lds_logits[R_ * BT_];   // [r][b]  72 KB
    __shared__ float lds_red[WAVES_ * 256];  // 8 KB scratch
    __shared__ float lds_out[256];           // out tile [m][n]
    __shared__ float lds_mz[2 * BT_];        // softmax max / 1/Z per batch

    const int tid  = threadIdx.x;
    const int lane = tid & 31;
    const int wv   = tid >> 5;
    const int half = (lane >> 4) & 1;
    const int bn   = lane & 15;          // A: batch row; B/D: cout column

    const int c  = blockIdx.y;
    const int b0 = blockIdx.x * BT_;

    const float* xb = x + ((size_t)(b0 + bn) * R_) * CIN_;
    const float* wc = rw + (size_t)c * R_ * CIN_ * COUT_;

    for (int idx = tid; idx < R_ * BT_; idx += 256) lds_logits[idx] = 0.0f;
    __syncthreads();

    // ---- iteration 0, weighted sum: logits==0 -> uniform probs 1/R ----
    v8f acc = zero8();
    for (int i = 0; i < RPW_; ++i) {
        const int r = wv * RPW_ + i;
        const float* xp = xb + (size_t)r * CIN_;
        const float* wp = wc + (size_t)r * (CIN_ * COUT_);
        __builtin_prefetch(wp + CIN_ * COUT_, 0, 3);   // global_prefetch_b8
        acc = wmma_k8(xp, wp, half, bn, 1.0f, acc);
    }
#pragma unroll
    for (int j = 0; j < 8; ++j) acc[j] *= (1.0f / (float)R_);
    reduce_squash(acc, lds_red, lds_out, tid, wv, half, bn);

    for (int iter = 0; iter < 2; ++iter) {
        // output tile in D layout for this lane
        v8f outv;
#pragma unroll
        for (int j = 0; j < 8; ++j) outv[j] = lds_out[(j + 8 * half) * 16 + bn];

        // ---- pass B: logits[b,r] += sum_o P[b,r,o] * out[b,o] ----
        for (int i = 0; i < RPW_; ++i) {
            const int r = wv * RPW_ + i;
            const float* xp = xb + (size_t)r * CIN_;
            const float* wp = wc + (size_t)r * (CIN_ * COUT_);
            __builtin_prefetch(wp + CIN_ * COUT_, 0, 3);
            v8f d = zero8();
            d = wmma_k8(xp, wp, half, bn, 1.0f, d);
            float t[8];
#pragma unroll
            for (int j = 0; j < 8; ++j) t[j] = d[j] * outv[j];
#pragma unroll
            for (int m = 1; m <= 8; m <<= 1) {
#pragma unroll
                for (int j = 0; j < 8; ++j) t[j] += __shfl_xor(t[j], m, 32);
            }
            if (bn == 0) {   // lanes 0 (b=0..7) and 16 (b=8..15) write
#pragma unroll
                for (int j = 0; j < 8; ++j)
                    lds_logits[r * BT_ + half * 8 + j] += t[j];
            }
        }
        __syncthreads();

        // ---- softmax stats over r per batch column ----
        const int bcol = tid & 15;
        const int rr0  = (tid >> 4) * (R_ / 16);   // 72 rows each
        float mloc = -3.0e38f;
        for (int k = 0; k < R_ / 16; ++k)
            mloc = fmaxf(mloc, lds_logits[(rr0 + k) * BT_ + bcol]);
        lds_red[tid] = mloc;
        __syncthreads();
        if (tid < BT_) {
            float m = -3.0e38f;
#pragma unroll
            for (int k = 0; k < 16; ++k) m = fmaxf(m, lds_red[tid + k * 16]);
            lds_mz[tid] = m;
        }
        __syncthreads();
        const float mb = lds_mz[bcol];
        float sloc = 0.0f;
        for (int k = 0; k < R_ / 16; ++k)
            sloc += __expf(lds_logits[(rr0 + k) * BT_ + bcol] - mb);
        lds_red[tid] = sloc;
        __syncthreads();
        if (tid < BT_) {
            float ssum = 0.0f;
#pragma unroll
            for (int k = 0; k < 16; ++k) ssum += lds_red[tid + k * 16];
            lds_mz[BT_ + tid] = 1.0f / ssum;
        }
        __syncthreads();

        // ---- pass A: S = sum_r softmax(logits) * P, probs folded into A ----
        const float mA  = lds_mz[bn];
        const float izA = lds_mz[BT_ + bn];
        acc = zero8();
        for (int i = 0; i < RPW_; ++i) {
            const int r = wv * RPW_ + i;
            const float p = __expf(lds_logits[r * BT_ + bn] - mA) * izA;
            const float* xp = xb + (size_t)r * CIN_;
            const float* wp = wc + (size_t)r * (CIN_ * COUT_);
            __builtin_prefetch(wp + CIN_ * COUT_, 0, 3);
            acc = wmma_k8(xp, wp, half, bn, p, acc);
        }
        reduce_squash(acc, lds_red, lds_out, tid, wv, half, bn);
    }

    // ---- write final outputs [caps][B][16] ----
    out[((size_t)c * B_ + (b0 + (tid >> 4))) * COUT_ + (tid & 15)] = lds_out[tid];
}

extern "C" void kernel_launch(void* const* d_in, const int* in_sizes, int n_in,
                              void* d_out, int out_size, void* d_ws, size_t ws_size,
                              hipStream_t stream) {
    const float* x  = (const float*)d_in[0];   // [256][1152][8]
    const float* rw = (const float*)d_in[1];   // [10][1152][8][16]
    float* out = (float*)d_out;                // [10][256][16]
    (void)in_sizes; (void)n_in; (void)out_size; (void)d_ws; (void)ws_size;

    dim3 grid(B_ / BT_, NCAPS_);               // 16 x 10 = 160 workgroups
    caps_route_wmma<<<grid, 256, 0, stream>>>(x, rw, out);
}